// TGNEncoder_68762426409422
// MI455X (gfx1250) — compile-verified
//
#include <hip/hip_runtime.h>
#include <hip/hip_bf16.h>
#include <math.h>
#include <stdint.h>

typedef __attribute__((ext_vector_type(2))) float v2f;
typedef __attribute__((ext_vector_type(8))) float v8f;

#define IN_DIM 128
#define HC     64      // H*C
#define NMAT   4
#define LDSN   132     // padded k-stride for transposed weights in LDS (conflict-free b64 reads)

// ---------------------------------------------------------------------------
// monotone float <-> uint encoding so unsigned atomicMax == float max
// ---------------------------------------------------------------------------
__device__ __forceinline__ unsigned enc_f32(float f) {
  unsigned u = __float_as_uint(f);
  return (u & 0x80000000u) ? ~u : (u | 0x80000000u);
}
__device__ __forceinline__ float dec_f32(unsigned u) {
  return (u & 0x80000000u) ? __uint_as_float(u ^ 0x80000000u) : __uint_as_float(~u);
}

// ---------------------------------------------------------------------------
// Fused 4-way GEMM:  Q|K|V|Skip = x @ W* + b*   using V_WMMA_F32_16X16X4_F32
// block = 256 threads (8 wave32), each wave computes a 16-row stripe of all
// four 64-wide outputs. Weights staged transposed in LDS (Wt[n][k], stride 132)
// via GLOBAL_LOAD_ASYNC_TO_LDS_B32 (per-lane LDS dest = free transpose,
// tracked on ASYNCcnt).
// ---------------------------------------------------------------------------
__global__ __launch_bounds__(256) void tgn_gemm_qkvs(
    const float* __restrict__ x,
    const float* __restrict__ Wq, const float* __restrict__ bq,
    const float* __restrict__ Wk, const float* __restrict__ bk,
    const float* __restrict__ Wv, const float* __restrict__ bv,
    const float* __restrict__ Ws, const float* __restrict__ bs,
    float* __restrict__ Q, float* __restrict__ K, float* __restrict__ V,
    float* __restrict__ Sout, int n)
{
  __shared__ float Wt[NMAT * HC * LDSN];   // 4 * 64 * 132 * 4B = 132 KB

  const float* Wsrc[NMAT] = {Wq, Wk, Wv, Ws};
  const float* Bsrc[NMAT] = {bq, bk, bv, bs};
  float*       Dst [NMAT] = {Q, K, V, Sout};

  const int tid = threadIdx.x;

  // stage weights transposed: Wt[mat][n][k] = W[mat][k*64 + n]
  // async copy: each lane scatters one element to its own LDS address
  for (int idx = tid; idx < NMAT * IN_DIM * HC; idx += 256) {
    int mat = idx >> 13;        // / 8192
    int r   = idx & 8191;       // r = k*64 + n
    int k   = r >> 6;
    int nn  = r & 63;
    unsigned lds_off = (unsigned)(uintptr_t)(&Wt[mat * (HC * LDSN) + nn * LDSN + k]);
    unsigned long long gaddr = (unsigned long long)(uintptr_t)(Wsrc[mat] + r);
    asm volatile("global_load_async_to_lds_b32 %0, %1, off"
                 :: "v"(lds_off), "v"(gaddr) : "memory");
  }
  asm volatile("s_wait_asynccnt 0x0" ::: "memory");
  __syncthreads();

  const int wave    = tid >> 5;
  const int lane    = tid & 31;
  const int m16     = lane & 15;
  const int khalf   = lane >> 4;                 // 0 or 1
  const int rowBase = blockIdx.x * 128 + wave * 16;
  const int rowA    = rowBase + m16;             // row this lane supplies for A
  const bool rowOK  = rowA < n;
  // clamp pointer for OOB lanes: load is always legal; garbage only reaches
  // WMMA output rows that are never stored (per-row A->D dependency)
  const float2* xrow = (const float2*)(x + (size_t)(rowOK ? rowA : 0) * IN_DIM);

  // accumulators: 4 matrices x 4 N-tiles of 16x16 fp32 (8 VGPRs each = 128 VGPRs)
  v8f acc[NMAT][4];
#pragma unroll
  for (int mat = 0; mat < NMAT; ++mat)
#pragma unroll
    for (int j = 0; j < 4; ++j) {
      float bval = Bsrc[mat][j * 16 + m16];      // bias depends on column only
      acc[mat][j] = (v8f){bval, bval, bval, bval, bval, bval, bval, bval};
    }

#pragma unroll 4
  for (int kk = 0; kk < IN_DIM; kk += 4) {
    const int kOff = kk + 2 * khalf;
    // A fragment 16x4 f32: lanes 0-15 -> K={kk,kk+1}, lanes 16-31 -> K={kk+2,kk+3}
    float2 af = xrow[kOff >> 1];                 // unconditional global_load_b64
    v2f a = {af.x, af.y};
#pragma unroll
    for (int mat = 0; mat < NMAT; ++mat) {
#pragma unroll
      for (int j = 0; j < 4; ++j) {
        // B fragment 4x16 f32: N = lane&15, same K split as A; ds_load_b64, bank-conflict-free
        const float2 bf = *(const float2*)&Wt[mat * (HC * LDSN) + (j * 16 + m16) * LDSN + kOff];
        v2f b = {bf.x, bf.y};
        acc[mat][j] = __builtin_amdgcn_wmma_f32_16x16x4_f32(
            false, a, false, b, (short)0, acc[mat][j], false, false);
      }
    }
  }

  // store: C/D layout -> VGPR r, lane l holds (row = r + 8*(l>>4), col = l&15)
#pragma unroll
  for (int mat = 0; mat < NMAT; ++mat) {
    float* D = Dst[mat];
#pragma unroll
    for (int j = 0; j < 4; ++j)
#pragma unroll
      for (int r = 0; r < 8; ++r) {
        int rr = rowBase + r + 8 * khalf;
        if (rr < n) D[(size_t)rr * HC + j * 16 + m16] = acc[mat][j][r];
      }
  }
}

// ---------------------------------------------------------------------------
// Pass B: per-edge logits + segment max (one wave32 per edge, lane = channel)
// ---------------------------------------------------------------------------
__global__ __launch_bounds__(256) void tgn_edge_logits(
    const int* __restrict__ src, const int* __restrict__ dst,
    const float* __restrict__ Q, const float* __restrict__ K,
    float* __restrict__ logits, unsigned* __restrict__ m_enc, int E)
{
  int e    = (int)((blockIdx.x * blockDim.x + threadIdx.x) >> 5);
  int lane = threadIdx.x & 31;
  if (e >= E) return;                      // warp-uniform exit
  int s = src[e], d = dst[e];
  const float* qd = Q + (size_t)d * HC;
  const float* ks = K + (size_t)s * HC;
  float p0 = qd[lane]      * ks[lane];        // head 0: channels 0..31
  float p1 = qd[lane + 32] * ks[lane + 32];   // head 1: channels 32..63
#pragma unroll
  for (int off = 16; off; off >>= 1) {
    p0 += __shfl_xor(p0, off, 32);
    p1 += __shfl_xor(p1, off, 32);
  }
  const float scale = 0.17677669529663687f;   // 1/sqrt(32)
  if (lane == 0) {
    float l0 = p0 * scale, l1 = p1 * scale;
    logits[(size_t)e * 2]     = l0;
    logits[(size_t)e * 2 + 1] = l1;
    atomicMax(&m_enc[(size_t)d * 2],     enc_f32(l0));
    atomicMax(&m_enc[(size_t)d * 2 + 1], enc_f32(l1));
  }
}

// ---------------------------------------------------------------------------
// Pass C: p = exp(logit - max); scatter-add p*v[src] and p (unnormalized softmax)
// ---------------------------------------------------------------------------
__global__ __launch_bounds__(256) void tgn_edge_accum(
    const int* __restrict__ src, const int* __restrict__ dst,
    const float* __restrict__ V, const float* __restrict__ logits,
    const unsigned* __restrict__ m_enc,
    float* __restrict__ acc, float* __restrict__ denom, int E)
{
  int e    = (int)((blockIdx.x * blockDim.x + threadIdx.x) >> 5);
  int lane = threadIdx.x & 31;
  if (e >= E) return;
  int s = src[e], d = dst[e];
  float l0 = logits[(size_t)e * 2];
  float l1 = logits[(size_t)e * 2 + 1];
  float m0 = dec_f32(m_enc[(size_t)d * 2]);
  float m1 = dec_f32(m_enc[(size_t)d * 2 + 1]);
  float p0 = __expf(l0 - m0);
  float p1 = __expf(l1 - m1);
  const float* vs = V + (size_t)s * HC;
  atomicAdd(&acc[(size_t)d * HC + lane],      p0 * vs[lane]);        // coalesced fadd
  atomicAdd(&acc[(size_t)d * HC + 32 + lane], p1 * vs[lane + 32]);
  if (lane == 0) {
    atomicAdd(&denom[(size_t)d * 2],     p0);
    atomicAdd(&denom[(size_t)d * 2 + 1], p1);
  }
}

// ---------------------------------------------------------------------------
// Pass D: out = skip (already in d_out) + acc / (denom + 1e-16)
// ---------------------------------------------------------------------------
__global__ __launch_bounds__(256) void tgn_finalize(
    const float* __restrict__ acc, const float* __restrict__ denom,
    float* __restrict__ out, int total)
{
  int i = blockIdx.x * blockDim.x + threadIdx.x;
  if (i >= total) return;
  out[i] += acc[i] / (denom[i >> 5] + 1e-16f);   // i>>5 == n*2 + head
}

// ---------------------------------------------------------------------------
extern "C" void kernel_launch(void* const* d_in, const int* in_sizes, int n_in,
                              void* d_out, int out_size, void* d_ws, size_t ws_size,
                              hipStream_t stream) {
  const int*   src = (const int*)d_in[0];
  const int*   dst = (const int*)d_in[1];
  // d_in[2] = t (unused by the math)
  const float* x   = (const float*)d_in[3];
  const float* Wq  = (const float*)d_in[4];
  const float* bq  = (const float*)d_in[5];
  const float* Wk  = (const float*)d_in[6];
  const float* bk  = (const float*)d_in[7];
  const float* Wv  = (const float*)d_in[8];
  const float* bv  = (const float*)d_in[9];
  const float* Wsk = (const float*)d_in[10];
  const float* bsk = (const float*)d_in[11];

  const int E = in_sizes[0];
  const int N = in_sizes[3] / IN_DIM;

  float* ws      = (float*)d_ws;
  float* Qb      = ws;                               // N*64
  float* Kb      = Qb  + (size_t)N * HC;             // N*64
  float* Vb      = Kb  + (size_t)N * HC;             // N*64
  float* accb    = Vb  + (size_t)N * HC;             // N*64
  float* logits  = accb + (size_t)N * HC;            // E*2
  float* denom   = logits + (size_t)E * 2;           // N*2
  unsigned* menc = (unsigned*)(denom + (size_t)N * 2); // N*2
  (void)menc;

  // clear accumulators / denom / running-max (capture-safe)
  hipMemsetAsync(accb, 0, (size_t)N * HC * sizeof(float), stream);
  hipMemsetAsync(denom, 0, (size_t)N * 4 * sizeof(float), stream); // denom + menc (adjacent)

  float* outp = (float*)d_out;

  tgn_gemm_qkvs<<<(N + 127) / 128, 256, 0, stream>>>(
      x, Wq, bq, Wk, bk, Wv, bv, Wsk, bsk, Qb, Kb, Vb, outp, N);

  int edgeBlocks = (E + 7) / 8;   // 8 wave32s per 256-thread block
  tgn_edge_logits<<<edgeBlocks, 256, 0, stream>>>(src, dst, Qb, Kb, logits, menc, E);
  tgn_edge_accum <<<edgeBlocks, 256, 0, stream>>>(src, dst, Vb, logits, menc, accb, denom, E);

  int total = N * HC;
  tgn_finalize<<<(total + 255) / 256, 256, 0, stream>>>(accb, denom, outp, total);
}